// DigitCaps_54649163874404
// MI455X (gfx1250) — compile-verified
//
#include <hip/hip_runtime.h>
#include <math.h>

typedef __attribute__((ext_vector_type(2))) float v2f;
typedef __attribute__((ext_vector_type(8))) float v8f;

#define R_TOTAL       6912
#define RSPLIT        27
#define ROUTES_PER_WG 256          // 6912 / 27
#define NWAVES        8
#define ROUTES_PER_WV 32           // 256 / 8
#define BT            4            // batch tiles of 16 (B = 64)
#define C_CAPS        10
#define NEG_INF       (-__builtin_inff())

// ---- Cross-lane 16-lane sum on the VALU pipe (no LDS traffic) -------------
// DPP8 selector constants: lane i reads lane i^k within its 8-lane group.
#define DPP8_XOR1 0x00DE54C1u   // [1,0,3,2,5,4,7,6]
#define DPP8_XOR2 0x00B3E21Au   // [2,3,0,1,6,7,4,5]
#define DPP8_XOR4 0x00688FACu   // [4,5,6,7,0,1,2,3]
#define DPP16_ROW_MIRROR 0x140  // lane n <- n^15 within 16-lane row

template <unsigned SEL>
__device__ __forceinline__ float dpp8_add(float v) {
  int p = __builtin_amdgcn_mov_dpp8(__float_as_int(v), SEL);
  return v + __int_as_float(p);
}

__device__ __forceinline__ float row_mirror_add(float v) {
  int p = __builtin_amdgcn_update_dpp(0, __float_as_int(v),
                                      DPP16_ROW_MIRROR, 0xF, 0xF, true);
  return v + __int_as_float(p);
}

// Sum across the 16 lanes of each half-wave: xor1/2/4 inside the 8-group,
// then ROW_MIRROR (xor15) folds the two 8-group sums together.
__device__ __forceinline__ float xor_add16(float v) {
  v = dpp8_add<DPP8_XOR1>(v);
  v = dpp8_add<DPP8_XOR2>(v);
  v = dpp8_add<DPP8_XOR4>(v);
  v = row_mirror_add(v);
  return v;
}

// All 8 D-rows at once: 8 independent DPP adds per step for full VALU ILP.
__device__ __forceinline__ void xor_add16x8(float (&t)[8]) {
#pragma unroll
  for (int j = 0; j < 8; ++j) t[j] = dpp8_add<DPP8_XOR1>(t[j]);
#pragma unroll
  for (int j = 0; j < 8; ++j) t[j] = dpp8_add<DPP8_XOR2>(t[j]);
#pragma unroll
  for (int j = 0; j < 8; ++j) t[j] = dpp8_add<DPP8_XOR4>(t[j]);
#pragma unroll
  for (int j = 0; j < 8; ++j) t[j] = row_mirror_add(t[j]);
}

// One routing pass over a 256-route chunk for one (c, batch-tile-of-16).
// Priors recomputed per route via V_WMMA_F32_16X16X4_F32 (K=8 as two K=4
// steps, full f32 precision, x/W stay L2-resident). WITH_L=false is the
// uniform-softmax iteration 0: no logits, plain prior accumulation.
template <bool WITH_L>
__global__ __launch_bounds__(256)
void caps_pass_kernel(const float* __restrict__ x,      // [B, R, 8]
                      const float* __restrict__ W,      // [C, R, 8, 16]
                      const float* __restrict__ Vcum,   // [C*BT, 16, 16]
                      float* __restrict__ partM,        // [C*BT*RSPLIT, 16]
                      float* __restrict__ partZ,        // [C*BT*RSPLIT, 16]
                      float* __restrict__ partS)        // [C*BT*RSPLIT, 16, 16]
{
  const int k    = blockIdx.x;          // route chunk
  const int bt   = blockIdx.y;          // batch tile
  const int c    = blockIdx.z;          // capsule class
  const int tid  = threadIdx.x;
  const int wave = tid >> 5;
  const int lane = tid & 31;
  const int half = lane >> 4;           // 0: batches 0-7 / K 0-1, 1: batches 8-15 / K 2-3
  const int n16  = lane & 15;
  const int cbt  = c * BT + bt;

  // Cumulative squash vector, lane-resident for the whole pass.
  // vc[j] = Vcum[cbt][b = j + 8*half][o = n16]  (matches the WMMA D layout)
  float vc[8];
  if (WITH_L) {
    const float* vb = Vcum + (long)cbt * 256;
#pragma unroll
    for (int j = 0; j < 8; ++j) vc[j] = vb[(j + 8 * half) * 16 + n16];
  }

  // Online-softmax accumulators (per D-row j): running max, Z, weighted sum.
  float m[8], Z[8], S[8];
#pragma unroll
  for (int j = 0; j < 8; ++j) { m[j] = NEG_INF; Z[j] = 0.0f; S[j] = 0.0f; }

  const int    rbase = k * ROUTES_PER_WG + wave * ROUTES_PER_WV;
  const long   xrow  = (long)(bt * 16 + n16) * R_TOTAL;      // batch row for A loads
  const float* wbase = W + (long)c * R_TOTAL * 128;

#pragma unroll 2
  for (int q = 0; q < ROUTES_PER_WV; ++q) {
    const int    r  = rbase + q;
    const float* xp = x + (xrow + r) * 8;
    const float* wp = wbase + (long)r * 128 + 32 * half + n16;

    // A (16x4 f32): VGPR0 = {K0 | K2}, VGPR1 = {K1 | K3} split at lane 16.
    v2f a0, a1;
    a0.x = xp[2 * half];     a0.y = xp[2 * half + 1];
    a1.x = xp[4 + 2 * half]; a1.y = xp[4 + 2 * half + 1];
    // B (4x16 f32), W[c,r] flat over (i,o): VGPR0 = {K0 | K2}, VGPR1 = {K1 | K3}.
    v2f b0, b1;
    b0.x = wp[0];  b0.y = wp[16];
    b1.x = wp[64]; b1.y = wp[80];

    // Keep the next route's W tile warm in the near cache.
    __builtin_prefetch(wp + 128, 0, 3);

    // P tile = x_tile[16b x 8i] * W[8i x 16o] via two chained K=4 WMMAs.
    v8f acc = {};
    acc = __builtin_amdgcn_wmma_f32_16x16x4_f32(false, a0, false, b0,
                                                (short)0, acc, false, false);
    acc = __builtin_amdgcn_wmma_f32_16x16x4_f32(false, a1, false, b1,
                                                (short)0, acc, false, false);

    if (WITH_L) {
      // logits on the fly: l[b] = sum_o P[b,o] * Vcum[b,o], all 8 rows batched
      float t[8];
#pragma unroll
      for (int j = 0; j < 8; ++j) t[j] = acc[j] * vc[j];
      xor_add16x8(t);
      // online softmax update
#pragma unroll
      for (int j = 0; j < 8; ++j) {
        const float lnew = t[j];
        const float mold = m[j];
        const float mnew = fmaxf(mold, lnew);
        const float cs   = __expf(mold - mnew);   // rescale old accumulators
        const float p    = __expf(lnew - mnew);
        m[j] = mnew;
        Z[j] = Z[j] * cs + p;
        S[j] = S[j] * cs + p * acc[j];
      }
    } else {
      // uniform probs: plain accumulation, fix up m/Z after the loop
#pragma unroll
      for (int j = 0; j < 8; ++j) S[j] += acc[j];
    }
  }

  if (!WITH_L) {
#pragma unroll
    for (int j = 0; j < 8; ++j) { m[j] = 0.0f; Z[j] = (float)ROUTES_PER_WV; }
  }

  // Merge the 8 waves' online-softmax states through LDS.
  __shared__ float wm[NWAVES][16];
  __shared__ float wz[NWAVES][16];
  __shared__ float wS[NWAVES][16][16];
#pragma unroll
  for (int j = 0; j < 8; ++j) {
    const int b = j + 8 * half;
    wS[wave][b][n16] = S[j];
    if (n16 == 0) { wm[wave][b] = m[j]; wz[wave][b] = Z[j]; }
  }
  __syncthreads();

  {
    const int b = tid >> 4, o = tid & 15;
    float M = NEG_INF;
#pragma unroll
    for (int w = 0; w < NWAVES; ++w) M = fmaxf(M, wm[w][b]);
    float Zt = 0.0f, St = 0.0f;
#pragma unroll
    for (int w = 0; w < NWAVES; ++w) {
      const float e = __expf(wm[w][b] - M);
      Zt += wz[w][b] * e;
      St += wS[w][b][o] * e;
    }
    const long pidx = (long)cbt * RSPLIT + k;
    partS[pidx * 256 + b * 16 + o] = St;
    if (o == 0) { partM[pidx * 16 + b] = M; partZ[pidx * 16 + b] = Zt; }
  }
}

// Combine chunk partials, normalize, squash, update Vcum, emit final output.
template <int ITER>
__global__ __launch_bounds__(256)
void caps_merge_kernel(const float* __restrict__ partM,
                       const float* __restrict__ partZ,
                       const float* __restrict__ partS,
                       float* __restrict__ Vcum,
                       float* __restrict__ out)        // [C, B, 1, 1, 16]
{
  const int bt  = blockIdx.x;
  const int c   = blockIdx.y;
  const int cbt = c * BT + bt;
  const int tid = threadIdx.x;
  const int b   = tid >> 4;     // lanes 0-15 / 16-31 of a wave share a batch
  const int o   = tid & 15;

  float M = NEG_INF;
  for (int k = 0; k < RSPLIT; ++k)
    M = fmaxf(M, partM[((long)cbt * RSPLIT + k) * 16 + b]);

  float Zt = 0.0f, St = 0.0f;
  for (int k = 0; k < RSPLIT; ++k) {
    const long pidx = (long)cbt * RSPLIT + k;
    const float e = __expf(partM[pidx * 16 + b] - M);
    Zt += partZ[pidx * 16 + b] * e;
    St += partS[pidx * 256 + b * 16 + o] * e;
  }

  const float s  = St / Zt;                       // softmax-weighted prior sum
  const float sq = xor_add16(s * s);              // ||s||^2 over O (16 lanes)
  const float v  = s * sq / ((1.0f + sq) * sqrtf(sq));   // squash

  const float vold = (ITER == 0) ? 0.0f : Vcum[(long)cbt * 256 + b * 16 + o];
  Vcum[(long)cbt * 256 + b * 16 + o] = vold + v;  // logits = P . Vcum next pass

  if (ITER == 2) {
    const int bglob = bt * 16 + b;
    out[((long)c * 64 + bglob) * 16 + o] = v;
  }
}

extern "C" void kernel_launch(void* const* d_in, const int* in_sizes, int n_in,
                              void* d_out, int out_size, void* d_ws, size_t ws_size,
                              hipStream_t stream) {
  const float* x = (const float*)d_in[0];   // [64, 6912, 8]
  const float* W = (const float*)d_in[1];   // [10, 6912, 8, 16]
  float* out = (float*)d_out;               // [10, 64, 1, 1, 16]

  float* ws    = (float*)d_ws;
  float* Vcum  = ws;                                   // 10*4*256      = 10240
  float* partM = Vcum  + (long)C_CAPS * BT * 256;      // 10*4*27*16    = 17280
  float* partZ = partM + (long)C_CAPS * BT * RSPLIT * 16;
  float* partS = partZ + (long)C_CAPS * BT * RSPLIT * 16;  // *256 = 276480
  // total ~1.25 MB of workspace

  const dim3 blk(256);
  const dim3 gridP(RSPLIT, BT, C_CAPS);   // 1080 workgroups
  const dim3 gridM(BT, C_CAPS);           // 40 workgroups

  // iteration 0: uniform softmax (no logits)
  caps_pass_kernel<false><<<gridP, blk, 0, stream>>>(x, W, Vcum, partM, partZ, partS);
  caps_merge_kernel<0><<<gridM, blk, 0, stream>>>(partM, partZ, partS, Vcum, out);
  // iteration 1
  caps_pass_kernel<true><<<gridP, blk, 0, stream>>>(x, W, Vcum, partM, partZ, partS);
  caps_merge_kernel<1><<<gridM, blk, 0, stream>>>(partM, partZ, partS, Vcum, out);
  // iteration 2: writes the output
  caps_pass_kernel<true><<<gridP, blk, 0, stream>>>(x, W, Vcum, partM, partZ, partS);
  caps_merge_kernel<2><<<gridM, blk, 0, stream>>>(partM, partZ, partS, Vcum, out);
}